// Set2SetPooling_15848429322417
// MI455X (gfx1250) — compile-verified
//
#include <hip/hip_runtime.h>
#include <hip/hip_bf16.h>
#include <math.h>

// ---------------------------------------------------------------------------
// Set2Set pooling for MI455X (gfx1250, wave32).
//   N=400000 nodes, D=256, B=2048 segments (index sorted -> contiguous runs),
//   T=3 steps.  HBM floor: 3 single passes over x (410MB) ~ 1.23GB ~ 53us.
//   LSTM gates + final projection use V_WMMA_F32_16X16X4_F32 (f32 matrix core).
//   Attention streams x via GLOBAL_LOAD_ASYNC_TO_LDS_B128 double buffering
//   (ASYNCcnt / s_wait_asynccnt), flash-style online segment softmax.
// ---------------------------------------------------------------------------

#define D      256
#define NB     2048
#define TWOD   512
#define FOURD  1024
#define CHUNK  32

typedef float v2f __attribute__((ext_vector_type(2)));
typedef float v8f __attribute__((ext_vector_type(8)));

// --------------------------- utility kernels -------------------------------

__global__ void zero_kernel(float* __restrict__ p, int n) {
  int i = blockIdx.x * blockDim.x + threadIdx.x;
  int stride = gridDim.x * blockDim.x;
  for (; i < n; i += stride) p[i] = 0.f;
}

__device__ __forceinline__ int lower_bound_ll(const long long* __restrict__ a,
                                              int n, long long v) {
  int lo = 0, hi = n;
  while (lo < hi) {
    int mid = (lo + hi) >> 1;
    if (a[mid] < v) lo = mid + 1; else hi = mid;
  }
  return lo;
}

// index is sorted ascending; find [start,end) of each segment (empty ok).
__global__ void seg_bounds_kernel(const long long* __restrict__ idx, int n,
                                  int* __restrict__ segs, int* __restrict__ sege) {
  int b = blockIdx.x * blockDim.x + threadIdx.x;
  if (b >= NB) return;
  segs[b] = lower_bound_ll(idx, n, (long long)b);
  sege[b] = lower_bound_ll(idx, n, (long long)b + 1);
}

// ------------------------------ WMMA GEMM ----------------------------------
// C[M,N] = A1[M,K1] * W1[N,K1]^T  (+ A2[M,K2] * W2[N,K2]^T) + bias1[N] (+bias2[N])
// One wave computes a 64x64 tile: 4x4 fragments of 16x16, K-step 4 (f32 WMMA).
//
// Fragment layouts (CDNA5 ISA 7.12.2, wave32):
//   A 16x4 : lane&15 = M row; reg r holds K = k + 2*(lane>>4) + r
//   B 4x16 : lane&15 = N col; reg r holds K = k + 2*(lane>>4) + r   (B=W^T)
//   C 16x16: lane&15 = N col; reg r holds M = m0 + r + 8*(lane>>4)

__global__ void __launch_bounds__(32)
gemm_wmma_f32(const float* __restrict__ A1, const float* __restrict__ W1, int K1,
              const float* __restrict__ A2, const float* __restrict__ W2, int K2,
              const float* __restrict__ bias1, const float* __restrict__ bias2,
              float* __restrict__ C, int M, int N) {
  const int lane = threadIdx.x;
  const int m0   = blockIdx.y * 64;
  const int n0   = blockIdx.x * 64;
  const int half = lane >> 4;       // 0 or 1
  const int koff = half * 2;
  const int r15  = lane & 15;

  v8f acc[4][4];
#pragma unroll
  for (int i = 0; i < 4; ++i)
#pragma unroll
    for (int j = 0; j < 4; ++j) acc[i][j] = (v8f)(0.f);

  // ---- pass 1: A1 * W1^T ----
  for (int k = 0; k < K1; k += 4) {
    v2f af[4], bf[4];
#pragma unroll
    for (int i = 0; i < 4; ++i) {
      const float* ap = A1 + (size_t)(m0 + i * 16 + r15) * K1 + k + koff;
      af[i].x = ap[0]; af[i].y = ap[1];
      const float* wp = W1 + (size_t)(n0 + i * 16 + r15) * K1 + k + koff;
      bf[i].x = wp[0]; bf[i].y = wp[1];
    }
#pragma unroll
    for (int i = 0; i < 4; ++i)
#pragma unroll
      for (int j = 0; j < 4; ++j)
        acc[i][j] = __builtin_amdgcn_wmma_f32_16x16x4_f32(
            false, af[i], false, bf[j], (short)0, acc[i][j], false, false);
  }

  // ---- pass 2 (optional): A2 * W2^T ----
  if (A2 != nullptr) {
    for (int k = 0; k < K2; k += 4) {
      v2f af[4], bf[4];
#pragma unroll
      for (int i = 0; i < 4; ++i) {
        const float* ap = A2 + (size_t)(m0 + i * 16 + r15) * K2 + k + koff;
        af[i].x = ap[0]; af[i].y = ap[1];
        const float* wp = W2 + (size_t)(n0 + i * 16 + r15) * K2 + k + koff;
        bf[i].x = wp[0]; bf[i].y = wp[1];
      }
#pragma unroll
      for (int i = 0; i < 4; ++i)
#pragma unroll
        for (int j = 0; j < 4; ++j)
          acc[i][j] = __builtin_amdgcn_wmma_f32_16x16x4_f32(
              false, af[i], false, bf[j], (short)0, acc[i][j], false, false);
    }
  }

  // ---- epilogue: bias + store ----
#pragma unroll
  for (int j = 0; j < 4; ++j) {
    const int col = n0 + j * 16 + r15;
    float bn = bias1[col];
    if (bias2 != nullptr) bn += bias2[col];
#pragma unroll
    for (int i = 0; i < 4; ++i) {
#pragma unroll
      for (int r = 0; r < 8; ++r) {
        const int row = m0 + i * 16 + r + 8 * half;
        C[(size_t)row * N + col] = acc[i][j][r] + bn;
      }
    }
  }
}

// ------------------------------ LSTM cell ----------------------------------
// gates laid out [B, 4D] in torch order i,f,g,o. One thread per (b,d).

__device__ __forceinline__ float sigmoidf_(float x) {
  return 1.f / (1.f + __expf(-x));
}

__global__ void __launch_bounds__(D)
lstm_update_kernel(const float* __restrict__ gates,
                   float* __restrict__ c, float* __restrict__ h) {
  const int b = blockIdx.x;
  const int d = threadIdx.x;
  const float* g = gates + (size_t)b * FOURD;
  const float gi = g[d];
  const float gf = g[D + d];
  const float gg = g[2 * D + d];
  const float go = g[3 * D + d];
  const size_t o = (size_t)b * D + d;
  const float cn = sigmoidf_(gf) * c[o] + sigmoidf_(gi) * tanhf(gg);
  c[o] = cn;
  h[o] = sigmoidf_(go) * tanhf(cn);
}

// --------------------- fused segment softmax-attention ---------------------
// One 256-thread workgroup per segment. Single pass over x using online
// (flash-style) softmax. Chunks of 32 nodes are staged HBM->LDS with CDNA5
// GLOBAL_LOAD_ASYNC_TO_LDS_B128 into a double buffer: wave issues next
// chunk's async copy, computes on current chunk, then s_wait_asynccnt 0 +
// barrier. Thread d owns r_acc[d]; dot products use stride-32 conflict-free
// LDS access + wave32 shuffle reduction.

__global__ void __launch_bounds__(D)
attn_kernel(const float* __restrict__ x,
            const int* __restrict__ segs, const int* __restrict__ sege,
            const float* __restrict__ h, float* __restrict__ q_star) {
  __shared__ __align__(16) float xs[2][CHUNK][D];   // 64 KB double buffer
  __shared__ float qs[D];
  __shared__ float es[CHUNK];

  const int b = blockIdx.x;
  const int d = threadIdx.x;
  qs[d] = h[(size_t)b * D + d];

  float racc = 0.f;
  float m = -INFINITY;
  float z = 0.f;

  const int s = segs[b];
  const int e = sege[b];
  const int wave = d >> 5;
  const int lane = d & 31;
  const int r4 = d >> 6;            // row-within-group-of-4 for staging
  const int c4 = (d & 63) * 4;      // float4 column (16B aligned)

  // issue async HBM->LDS copies for rows [base, base+cnt) into xs[buf]
  auto stage = [&](int base, int cnt, int buf) {
#pragma unroll
    for (int i0 = 0; i0 < CHUNK; i0 += 4) {
      const int i = i0 + r4;
      if (i < cnt) {
        const unsigned long long gaddr =
            (unsigned long long)(size_t)&x[(size_t)(base + i) * D + c4];
        const unsigned lds_off = (unsigned)(size_t)&xs[buf][i][c4];
        asm volatile("global_load_async_to_lds_b128 %0, %1, off"
                     :: "v"(lds_off), "v"(gaddr) : "memory");
      }
    }
  };

  int buf = 0;
  if (s < e) stage(s, min(CHUNK, e - s), 0);
  asm volatile("s_wait_asynccnt 0x0" ::: "memory");
  __syncthreads();

  for (int base = s; base < e; base += CHUNK) {
    const int cnt = min(CHUNK, e - base);

    // prefetch next chunk into the other buffer (its readers finished at the
    // barrier that ended the previous iteration)
    const int nbase = base + CHUNK;
    if (nbase < e) stage(nbase, min(CHUNK, e - nbase), buf ^ 1);

    // e_i = <x_i, q>: wave w handles rows w, w+8, ...; stride-32 LDS access
    // keeps all 32 lanes on distinct banks.
    for (int i = wave; i < cnt; i += 8) {
      float p = 0.f;
#pragma unroll
      for (int j = 0; j < 8; ++j)
        p += xs[buf][i][lane + 32 * j] * qs[lane + 32 * j];
#pragma unroll
      for (int off = 16; off > 0; off >>= 1) p += __shfl_down(p, off, 32);
      if (lane == 0) es[i] = p;
    }
    __syncthreads();

    // online softmax update (each thread redundantly tracks m,z; owns dim d)
    float mc = -INFINITY;
    for (int i = 0; i < cnt; ++i) mc = fmaxf(mc, es[i]);
    const float nm = fmaxf(m, mc);
    const float sc = __expf(m - nm);        // exp(-inf)=0 on first chunk
    z *= sc;
    racc *= sc;
    for (int i = 0; i < cnt; ++i) {
      const float w = __expf(es[i] - nm);
      z += w;
      racc += w * xs[buf][i][d];
    }
    m = nm;

    // next buffer ready + es/xs reads retired before anyone proceeds
    asm volatile("s_wait_asynccnt 0x0" ::: "memory");
    __syncthreads();
    buf ^= 1;
  }

  const float r = (z > 0.f) ? (racc / z) : 0.f;   // empty segment -> 0
  q_star[(size_t)b * TWOD + d] = qs[d];           // q = h
  q_star[(size_t)b * TWOD + D + d] = r;
}

// ------------------------------ orchestration ------------------------------

extern "C" void kernel_launch(void* const* d_in, const int* in_sizes, int n_in,
                              void* d_out, int out_size, void* d_ws, size_t ws_size,
                              hipStream_t stream) {
  const float*     x     = (const float*)d_in[0];       // [N, 256]
  const long long* index = (const long long*)d_in[1];   // [N] sorted int64
  const float*     Wih   = (const float*)d_in[2];       // [1024, 512]
  const float*     Whh   = (const float*)d_in[3];       // [1024, 256]
  const float*     bih   = (const float*)d_in[4];       // [1024]
  const float*     bhh   = (const float*)d_in[5];       // [1024]
  const float*     Wp    = (const float*)d_in[6];       // [256, 512]
  const float*     bp    = (const float*)d_in[7];       // [256]
  float*           out   = (float*)d_out;               // [2048, 256]
  const int n = in_sizes[1];                             // node count

  // workspace layout (16MB floats + seg bounds)
  float* h      = (float*)d_ws;                 // [2048, 256]
  float* c      = h + (size_t)NB * D;           // [2048, 256]
  float* q_star = c + (size_t)NB * D;           // [2048, 512]
  float* gates  = q_star + (size_t)NB * TWOD;   // [2048, 1024]
  int*   segs   = (int*)(gates + (size_t)NB * FOURD);
  int*   sege   = segs + NB;

  // deterministic init each call (harness does not re-zero between replays)
  zero_kernel<<<512, 256, 0, stream>>>(h, NB * (D + D + TWOD));
  seg_bounds_kernel<<<(NB + 255) / 256, 256, 0, stream>>>(index, n, segs, sege);

  for (int t = 0; t < 3; ++t) {
    // gates = q_star @ Wih^T + h @ Whh^T + bih + bhh
    gemm_wmma_f32<<<dim3(FOURD / 64, NB / 64), 32, 0, stream>>>(
        q_star, Wih, TWOD, h, Whh, D, bih, bhh, gates, NB, FOURD);
    lstm_update_kernel<<<NB, D, 0, stream>>>(gates, c, h);
    attn_kernel<<<NB, D, 0, stream>>>(x, segs, sege, h, q_star);
  }

  // out = q_star @ Wp^T + bp
  gemm_wmma_f32<<<dim3(D / 64, NB / 64), 32, 0, stream>>>(
      q_star, Wp, TWOD, nullptr, nullptr, 0, bp, nullptr, out, NB, D);
}